// GNNModel_33758442946626
// MI455X (gfx1250) — compile-verified
//
#include <hip/hip_runtime.h>
#include <cstdint>

typedef __attribute__((ext_vector_type(2))) float v2f;
typedef __attribute__((ext_vector_type(8))) float v8f;

#define CFEAT 128   // C_IN == C_HID

// ---------------- graph preprocessing ----------------

__global__ void k_init(int* __restrict__ deg, int* __restrict__ cursor, int n) {
    int i = blockIdx.x * blockDim.x + threadIdx.x;
    if (i < n) { deg[i] = 0; cursor[i] = 0; }
}

__global__ void k_count(const int* __restrict__ dst, int* __restrict__ deg, int e) {
    int i = blockIdx.x * blockDim.x + threadIdx.x;
    if (i < e) atomicAdd(&deg[dst[i]], 1);
}

__global__ void k_dinv(const int* __restrict__ deg, float* __restrict__ dinv, int n) {
    int i = blockIdx.x * blockDim.x + threadIdx.x;
    // self-loop adds 1 to every node's degree -> always > 0
    if (i < n) dinv[i] = rsqrtf((float)deg[i] + 1.0f);
}

// single-workgroup exclusive scan (n ~ 50k, negligible vs rest of pipeline)
__global__ __launch_bounds__(1024)
void k_scan(const int* __restrict__ deg, int* __restrict__ offs, int n) {
    __shared__ int buf[1024];
    __shared__ int carry;
    const int tid = threadIdx.x;
    if (tid == 0) carry = 0;
    __syncthreads();
    for (int base = 0; base < n; base += 1024) {
        int i = base + tid;
        int v = (i < n) ? deg[i] : 0;
        buf[tid] = v;
        __syncthreads();
        #pragma unroll
        for (int off = 1; off < 1024; off <<= 1) {
            int t = (tid >= off) ? buf[tid - off] : 0;
            __syncthreads();
            buf[tid] += t;
            __syncthreads();
        }
        int incl = buf[tid];
        int c = carry;
        if (i < n) offs[i] = c + incl - v;   // exclusive
        __syncthreads();
        if (tid == 0) carry = c + buf[1023];
        __syncthreads();
    }
    if (tid == 0) offs[n] = carry;
}

__global__ void k_fill(const int* __restrict__ src, const int* __restrict__ dst,
                       const int* __restrict__ offs, int* __restrict__ cursor,
                       int* __restrict__ csr, int e) {
    int i = blockIdx.x * blockDim.x + threadIdx.x;
    if (i < e) {
        int d = dst[i];
        int p = atomicAdd(&cursor[d], 1);
        csr[offs[d] + p] = src[i];
    }
}

// ---------------- h = x @ W1 via V_WMMA_F32_16X16X4_F32 ----------------
// Per wave: 16 output rows x 128 output cols (8 col tiles), K=128 in steps of 4.
// A layout (ISA 32-bit A 16x4): lanes 0-15 -> K = k,k+1 ; lanes 16-31 -> K = k+2,k+3.
// B mirrored (4x16): VGPR0 = row k+2*hi, VGPR1 = row k+1+2*hi, col = lane&15.
// C/D layout: VGPR r, lane L -> (M = r + 8*(L>=16), N = L&15).

__global__ __launch_bounds__(256)
void k_gemm1(const float* __restrict__ x, const float* __restrict__ W1,
             float* __restrict__ h, int n) {
    __shared__ float sW[CFEAT * CFEAT];   // 64 KB
    for (int i = threadIdx.x; i < (CFEAT * CFEAT) / 4; i += 256)
        ((float4*)sW)[i] = ((const float4*)W1)[i];
    __syncthreads();

    const int lane = threadIdx.x & 31;
    const int wave = threadIdx.x >> 5;
    const int hi   = lane >> 4;      // 0: lanes 0-15, 1: lanes 16-31
    const int lo   = lane & 15;
    const int rowbase = (blockIdx.x * 8 + wave) * 16;
    if (rowbase >= n) return;        // wave-uniform

    int arow = rowbase + lo;
    if (arow >= n) arow = n - 1;     // clamp tail (stores guarded below)
    const float* __restrict__ xrow = x + (size_t)arow * CFEAT;

    v8f acc[8];
    const v8f z = {0.f, 0.f, 0.f, 0.f, 0.f, 0.f, 0.f, 0.f};
    #pragma unroll
    for (int t = 0; t < 8; ++t) acc[t] = z;

    for (int k = 0; k < CFEAT; k += 4) {
        v2f a;
        a.x = xrow[k + 2 * hi];
        a.y = xrow[k + 2 * hi + 1];
        #pragma unroll
        for (int t = 0; t < 8; ++t) {
            v2f b;
            const int col = t * 16 + lo;
            b.x = sW[(k + 2 * hi) * CFEAT + col];
            b.y = sW[(k + 2 * hi + 1) * CFEAT + col];
            acc[t] = __builtin_amdgcn_wmma_f32_16x16x4_f32(
                false, a, false, b, (short)0, acc[t], false, false);
        }
    }

    if (rowbase + 16 <= n) {
        // fast path: full tile, unguarded coalesced stores
        #pragma unroll
        for (int t = 0; t < 8; ++t) {
            #pragma unroll
            for (int r = 0; r < 8; ++r) {
                const int row = rowbase + r + 8 * hi;
                h[(size_t)row * CFEAT + t * 16 + lo] = acc[t][r];
            }
        }
    } else {
        #pragma unroll
        for (int t = 0; t < 8; ++t) {
            #pragma unroll
            for (int r = 0; r < 8; ++r) {
                const int row = rowbase + r + 8 * hi;
                if (row < n) h[(size_t)row * CFEAT + t * 16 + lo] = acc[t][r];
            }
        }
    }
}

// ---------------- propagation: wave per dst node, lane = 4 channels ----------------
// Accumulates sum_{src} h[src]*dinv[src] (+ h[node]*dinv[node] for the self loop),
// then scales once by dinv[node] at the end.

__device__ __forceinline__ float4 aggregate(const float* __restrict__ hsrc,
                                            const int* __restrict__ offs,
                                            const int* __restrict__ csr,
                                            const float* __restrict__ dinv,
                                            int node, int lane) {
    const float din = dinv[node];
    const float4* __restrict__ h4 = (const float4*)hsrc;
    float4 acc;
    {   // self loop contributes h[node]*din (scaled by din again at the end)
        float4 v = h4[(size_t)node * 32 + lane];
        acc.x = v.x * din; acc.y = v.y * din; acc.z = v.z * din; acc.w = v.w * din;
    }
    const int beg = offs[node], end = offs[node + 1];
    for (int j = beg; j < end; j += 32) {
        const int idx = j + lane;
        const int   s  = (idx < end) ? csr[idx] : 0;
        const float ws = (idx < end) ? dinv[s] : 0.0f;
        int cnt = end - j; if (cnt > 32) cnt = 32;
        int t = 0;
        if (cnt == 32) {
            // full chunk: 4 independent b128 gathers in flight per iteration
            for (; t < 32; t += 4) {
                const int   s0 = __shfl(s, t,     32), s1 = __shfl(s, t + 1, 32);
                const int   s2 = __shfl(s, t + 2, 32), s3 = __shfl(s, t + 3, 32);
                const float w0 = __shfl(ws, t,     32), w1 = __shfl(ws, t + 1, 32);
                const float w2 = __shfl(ws, t + 2, 32), w3 = __shfl(ws, t + 3, 32);
                float4 v0 = h4[(size_t)s0 * 32 + lane];
                float4 v1 = h4[(size_t)s1 * 32 + lane];
                float4 v2 = h4[(size_t)s2 * 32 + lane];
                float4 v3 = h4[(size_t)s3 * 32 + lane];
                acc.x += v0.x * w0; acc.y += v0.y * w0; acc.z += v0.z * w0; acc.w += v0.w * w0;
                acc.x += v1.x * w1; acc.y += v1.y * w1; acc.z += v1.z * w1; acc.w += v1.w * w1;
                acc.x += v2.x * w2; acc.y += v2.y * w2; acc.z += v2.z * w2; acc.w += v2.w * w2;
                acc.x += v3.x * w3; acc.y += v3.y * w3; acc.z += v3.z * w3; acc.w += v3.w * w3;
            }
        }
        for (; t < cnt; ++t) {
            const int   sk = __shfl(s, t, 32);
            const float wk = __shfl(ws, t, 32);
            float4 v = h4[(size_t)sk * 32 + lane];
            acc.x += v.x * wk; acc.y += v.y * wk; acc.z += v.z * wk; acc.w += v.w * wk;
        }
    }
    acc.x *= din; acc.y *= din; acc.z *= din; acc.w *= din;
    return acc;
}

__global__ __launch_bounds__(256)
void k_prop1(const float* __restrict__ h, float* __restrict__ h2,
             const int* __restrict__ offs, const int* __restrict__ csr,
             const float* __restrict__ dinv, const float* __restrict__ b1, int n) {
    const int node = (int)((blockIdx.x * blockDim.x + threadIdx.x) >> 5);
    const int lane = threadIdx.x & 31;
    if (node >= n) return;
    float4 acc = aggregate(h, offs, csr, dinv, node, lane);
    const float4 b = ((const float4*)b1)[lane];
    acc.x = fmaxf(acc.x + b.x, 0.0f);
    acc.y = fmaxf(acc.y + b.y, 0.0f);
    acc.z = fmaxf(acc.z + b.z, 0.0f);
    acc.w = fmaxf(acc.w + b.w, 0.0f);
    ((float4*)h2)[(size_t)node * 32 + lane] = acc;
}

// second propagation fused with the [128 -> 4x2] output projection:
// out[m, node, o] = sum_f agg[node, f] * W_out[m, f, o] + b_out[m, o]
__global__ __launch_bounds__(256)
void k_prop2_out(const float* __restrict__ h2, float* __restrict__ out,
                 const int* __restrict__ offs, const int* __restrict__ csr,
                 const float* __restrict__ dinv, const float* __restrict__ Wout,
                 const float* __restrict__ bout, int n) {
    const int node = (int)((blockIdx.x * blockDim.x + threadIdx.x) >> 5);
    const int lane = threadIdx.x & 31;
    if (node >= n) return;

    // per-lane W_out slice: f = lane*4 + j
    float wl[4][4][2];
    #pragma unroll
    for (int m = 0; m < 4; ++m)
        #pragma unroll
        for (int j = 0; j < 4; ++j) {
            const float2 w = ((const float2*)Wout)[m * CFEAT + lane * 4 + j];
            wl[m][j][0] = w.x; wl[m][j][1] = w.y;
        }

    float4 acc = aggregate(h2, offs, csr, dinv, node, lane);

    float p[4][2];
    #pragma unroll
    for (int m = 0; m < 4; ++m)
        #pragma unroll
        for (int o = 0; o < 2; ++o)
            p[m][o] = acc.x * wl[m][0][o] + acc.y * wl[m][1][o] +
                      acc.z * wl[m][2][o] + acc.w * wl[m][3][o];

    // wave32 reduction over channels
    #pragma unroll
    for (int m = 0; m < 4; ++m)
        #pragma unroll
        for (int o = 0; o < 2; ++o) {
            float v = p[m][o];
            #pragma unroll
            for (int d = 16; d > 0; d >>= 1) v += __shfl_xor(v, d, 32);
            p[m][o] = v;
        }

    if (lane == 0) {
        #pragma unroll
        for (int m = 0; m < 4; ++m) {
            float2 r;
            r.x = p[m][0] + bout[m * 2 + 0];
            r.y = p[m][1] + bout[m * 2 + 1];
            ((float2*)out)[(size_t)m * n + node] = r;
        }
    }
}

// ---------------- host launcher ----------------

extern "C" void kernel_launch(void* const* d_in, const int* in_sizes, int n_in,
                              void* d_out, int out_size, void* d_ws, size_t ws_size,
                              hipStream_t stream) {
    const float* x    = (const float*)d_in[0];
    const int*   ei   = (const int*)d_in[1];
    const float* W1   = (const float*)d_in[2];
    const float* b1   = (const float*)d_in[3];
    const float* Wout = (const float*)d_in[4];
    const float* bout = (const float*)d_in[5];
    float* out = (float*)d_out;

    const int n = in_sizes[0] / CFEAT;
    const int e = in_sizes[1] / 2;
    const int* srcv = ei;
    const int* dstv = ei + e;

    uintptr_t p = (uintptr_t)d_ws;
    auto alloc = [&](size_t bytes) {
        p = (p + 15) & ~(uintptr_t)15;
        void* r = (void*)p;
        p += bytes;
        return r;
    };
    int*   deg    = (int*)  alloc((size_t)n * 4);
    int*   cursor = (int*)  alloc((size_t)n * 4);
    int*   offs   = (int*)  alloc((size_t)(n + 1) * 4);
    int*   csr    = (int*)  alloc((size_t)e * 4);
    float* dinv   = (float*)alloc((size_t)n * 4);
    float* h      = (float*)alloc((size_t)n * CFEAT * 4);
    float* h2     = (float*)alloc((size_t)n * CFEAT * 4);
    (void)ws_size; (void)n_in; (void)out_size;

    const int TB = 256;
    const int gn = (n + TB - 1) / TB;
    const int ge = (e + TB - 1) / TB;

    k_init <<<gn, TB, 0, stream>>>(deg, cursor, n);
    k_count<<<ge, TB, 0, stream>>>(dstv, deg, e);
    k_dinv <<<gn, TB, 0, stream>>>(deg, dinv, n);
    k_scan <<<1, 1024, 0, stream>>>(deg, offs, n);
    k_fill <<<ge, TB, 0, stream>>>(srcv, dstv, offs, cursor, csr, e);

    k_gemm1<<<(n + 127) / 128, TB, 0, stream>>>(x, W1, h, n);

    const int gprop = (n + 7) / 8;   // 8 waves (nodes) per 256-thread block
    k_prop1    <<<gprop, TB, 0, stream>>>(h, h2, offs, csr, dinv, b1, n);
    k_prop2_out<<<gprop, TB, 0, stream>>>(h2, out, offs, csr, dinv, Wout, bout, n);
}